// RecurrentGCN_33457795236810
// MI455X (gfx1250) — compile-verified
//
#include <hip/hip_runtime.h>
#include <math.h>

#define F 128

typedef __attribute__((ext_vector_type(2))) float v2f;
typedef __attribute__((ext_vector_type(8))) float v8f;
typedef __attribute__((ext_vector_type(4))) int   v4i;

#if defined(__has_builtin)
#  if __has_builtin(__builtin_amdgcn_global_load_async_to_lds_b128) && \
      __has_builtin(__builtin_amdgcn_s_wait_asynccnt)
#    define USE_ASYNC_LDS 1
#  endif
#endif

// Stage 16 bytes global -> LDS. Async (ASYNCcnt, bypasses VGPRs) when available.
__device__ __forceinline__ void stage16B(const float* __restrict__ g, float* l) {
#ifdef USE_ASYNC_LDS
  __builtin_amdgcn_global_load_async_to_lds_b128(
      (__attribute__((address_space(1))) v4i*)g,
      (__attribute__((address_space(3))) v4i*)l, 0, 0);
#else
  *(float4*)l = *(const float4*)g;
#endif
}
__device__ __forceinline__ void stage_wait() {
#ifdef USE_ASYNC_LDS
  __builtin_amdgcn_s_wait_asynccnt(0);
#endif
}

// ---------------- p-norm (single wave) ----------------
__global__ void pnorm_kernel(const float* __restrict__ p, float* __restrict__ pn) {
  int lane = threadIdx.x;
  float s = 0.f;
  for (int k = lane; k < F; k += 32) { float v = p[k]; s += v * v; }
  #pragma unroll
  for (int m = 16; m > 0; m >>= 1) s += __shfl_xor(s, m, 32);
  if (lane == 0) *pn = sqrtf(s);
}

// ---------------- scores: one wave per row ----------------
__global__ void score_kernel(const float* __restrict__ x, const float* __restrict__ p,
                             const float* __restrict__ pn, float* __restrict__ scores, int n) {
  int tid  = blockIdx.x * blockDim.x + threadIdx.x;
  int row  = tid >> 5, lane = tid & 31;
  if (row >= n) return;
  const float4* xr = (const float4*)(x + (size_t)row * F);
  const float4* p4 = (const float4*)p;
  float4 a = xr[lane], b = p4[lane];
  float s = a.x * b.x + a.y * b.y + a.z * b.z + a.w * b.w;
  #pragma unroll
  for (int m = 16; m > 0; m >>= 1) s += __shfl_xor(s, m, 32);
  if (lane == 0) scores[row] = tanhf(s / *pn);
}

// ---------------- exact top-128 (stable, descending), single block ----------------
__global__ __launch_bounds__(1024) void topk_kernel(const float* __restrict__ scores, int n,
                                                    int* __restrict__ topidx, float* __restrict__ topval) {
  const int t = threadIdx.x;
  float local[128];                     // private copy of this thread's strided slice
  int cnt = 0;
  for (int i = t; i < n && cnt < 128; i += 1024) local[cnt++] = scores[i];
  __shared__ float sval[1024];
  __shared__ int   sidx[1024];
  for (int k = 0; k < 128; ++k) {
    float best = -INFINITY; int bi = 0x7fffffff;
    for (int c = 0; c < cnt; ++c) {
      float v = local[c];
      if (v > best) { best = v; bi = t + c * 1024; }   // ascending c => smallest index wins ties
    }
    sval[t] = best; sidx[t] = bi;
    __syncthreads();
    for (int s = 512; s > 0; s >>= 1) {
      if (t < s) {
        float v2 = sval[t + s]; int i2 = sidx[t + s];
        if (v2 > sval[t] || (v2 == sval[t] && i2 < sidx[t])) { sval[t] = v2; sidx[t] = i2; }
      }
      __syncthreads();
    }
    int win = sidx[0];
    if (t == 0) { topidx[k] = win; topval[k] = sval[0]; }
    if ((win & 1023) == t) local[win >> 10] = -INFINITY;   // owner masks it out
    __syncthreads();
  }
}

// ---------------- x_tilde = x[idx] * top_scores ----------------
__global__ void gather_kernel(const float* __restrict__ x, const int* __restrict__ topidx,
                              const float* __restrict__ topval, float* __restrict__ xt) {
  int k = blockIdx.x, j = threadIdx.x;
  xt[k * F + j] = x[(size_t)topidx[k] * F + j] * topval[k];
}

// ---------------- GRU step -> evolved weight W (128x128) ----------------
__global__ void gru_kernel(const float* __restrict__ xt, const float* __restrict__ w0,
                           const float* __restrict__ w_ih, const float* __restrict__ w_hh,
                           const float* __restrict__ b_ih, const float* __restrict__ b_hh,
                           float* __restrict__ Wm) {
  int i = blockIdx.x, j = threadIdx.x;
  __shared__ float xr[F], hr[F];
  xr[j] = xt[i * F + j];
  hr[j] = w0[i * F + j];
  __syncthreads();
  float ir = b_ih[j], iz = b_ih[F + j], in_ = b_ih[2 * F + j];
  float gr = b_hh[j], gz = b_hh[F + j], gn  = b_hh[2 * F + j];
  for (int k = 0; k < F; ++k) {
    float xv = xr[k], hv = hr[k];
    ir = fmaf(xv, w_ih[(size_t)j * F + k], ir);
    iz = fmaf(xv, w_ih[(size_t)(F + j) * F + k], iz);
    in_ = fmaf(xv, w_ih[(size_t)(2 * F + j) * F + k], in_);
    gr = fmaf(hv, w_hh[(size_t)j * F + k], gr);
    gz = fmaf(hv, w_hh[(size_t)(F + j) * F + k], gz);
    gn = fmaf(hv, w_hh[(size_t)(2 * F + j) * F + k], gn);
  }
  float r = 1.f / (1.f + expf(-(ir + gr)));
  float z = 1.f / (1.f + expf(-(iz + gz)));
  float nn = tanhf(in_ + r * gn);
  Wm[i * F + j] = (1.f - z) * nn + z * hr[j];
}

// ---------------- degree accumulation ----------------
__global__ void zero_kernel(float* __restrict__ buf, int n) {
  int i = blockIdx.x * blockDim.x + threadIdx.x;
  if (i < n) buf[i] = 0.f;
}

__global__ void deg_kernel(const int* __restrict__ ei, const float* __restrict__ ew,
                           float* __restrict__ deg, int E, int n) {
  long long tid = (long long)blockIdx.x * blockDim.x + threadIdx.x;
  if (tid < E) {
    unsafeAtomicAdd(&deg[ei[(size_t)E + tid]], ew[tid]);
  } else if (tid < (long long)E + n) {
    unsafeAtomicAdd(&deg[tid - E], 1.0f);   // self loop weight 1
  }
}

__global__ void dinv_kernel(const float* __restrict__ deg, float* __restrict__ dinv, int n) {
  int i = blockIdx.x * blockDim.x + threadIdx.x;
  if (i < n) { float d = deg[i]; dinv[i] = d > 0.f ? rsqrtf(d) : 0.f; }
}

// ---------------- xw = x @ W via V_WMMA_F32_16X16X4_F32 ----------------
// block = 256 threads = 8 waves; block handles 16 rows x 128 cols; wave w -> cols [16w,16w+16)
__global__ __launch_bounds__(256) void gemm_xw_kernel(const float* __restrict__ x,
                                                      const float* __restrict__ Wm,
                                                      float* __restrict__ xw) {
  __shared__ float As[16 * 132];
  const int m0 = blockIdx.x * 16;
  const int t  = threadIdx.x;
  {
    int r = t >> 4, c = (t & 15) * 8;
    const float* g = x + (size_t)(m0 + r) * F + c;
    float* l = &As[r * 132 + c];
    stage16B(g, l);
    stage16B(g + 4, l + 4);
  }
  stage_wait();
  __syncthreads();
  const int lane = t & 31, khalf = lane >> 4, mn = lane & 15;
  const int n0 = (t >> 5) * 16;
  v8f acc = {};
  #pragma unroll
  for (int k0 = 0; k0 < F; k0 += 4) {
    int kb = k0 + 2 * khalf;
    v2f a, b;
    a.x = As[mn * 132 + kb];
    a.y = As[mn * 132 + kb + 1];
    b.x = Wm[(size_t)kb * F + n0 + mn];
    b.y = Wm[(size_t)(kb + 1) * F + n0 + mn];
    acc = __builtin_amdgcn_wmma_f32_16x16x4_f32(false, a, false, b, (short)0, acc, false, false);
  }
  #pragma unroll
  for (int r = 0; r < 8; ++r)
    xw[(size_t)(m0 + r + 8 * khalf) * F + n0 + mn] = acc[r];
}

// ---------------- agg init = self-loop term: dinv[i]^2 * xw[i,:] ----------------
__global__ void agginit_kernel(const float* __restrict__ xw, const float* __restrict__ dinv,
                               float* __restrict__ agg, int n) {
  size_t tid = (size_t)blockIdx.x * blockDim.x + threadIdx.x;
  if (tid >= (size_t)n * F) return;
  int row = (int)(tid >> 7);
  float d = dinv[row];
  agg[tid] = d * d * xw[tid];
}

// ---------------- edge scatter: one wave per edge, float4 per lane ----------------
// Per wave: ~5 scalar loads total (lane-split + shfl broadcast), one coalesced 512B
// row gather from L2-resident xw, 128 global_atomic_add_f32 into L2-resident agg.
__global__ void edge_kernel(const int* __restrict__ ei, const float* __restrict__ ew,
                            const float* __restrict__ dinv, const float* __restrict__ xw,
                            float* __restrict__ agg, int E) {
  long long tid = (long long)blockIdx.x * blockDim.x + threadIdx.x;
  long long e = tid >> 5;
  int lane = (int)(tid & 31);
  if (e >= E) return;                       // wave-uniform guard (e shared by all 32 lanes)
  int sd = 0;
  if (lane < 2) sd = ei[(size_t)lane * E + e];   // lane0: src, lane1: dst
  int s = __shfl(sd, 0, 32);
  int d = __shfl(sd, 1, 32);
  float nrm = 0.f;
  if (lane == 0) nrm = dinv[s] * ew[e] * dinv[d];
  nrm = __shfl(nrm, 0, 32);
  float4 v = *(const float4*)(xw + (size_t)s * F + lane * 4);
  float* o = agg + (size_t)d * F + lane * 4;
  unsafeAtomicAdd(o + 0, nrm * v.x);
  unsafeAtomicAdd(o + 1, nrm * v.y);
  unsafeAtomicAdd(o + 2, nrm * v.z);
  unsafeAtomicAdd(o + 3, nrm * v.w);
}

// ---------------- out = relu(agg) @ w_lin^T + b_lin (WMMA f32) ----------------
__global__ __launch_bounds__(256) void gemm_out_kernel(const float* __restrict__ agg,
                                                       const float* __restrict__ wlin,
                                                       const float* __restrict__ blin,
                                                       float* __restrict__ out) {
  __shared__ float As[16 * 132];
  const int m0 = blockIdx.x * 16;
  const int t  = threadIdx.x;
  {
    int r = t >> 4, c = (t & 15) * 8;
    const float* g = agg + (size_t)(m0 + r) * F + c;
    float* l = &As[r * 132 + c];
    stage16B(g, l);          // stage raw; relu applied on fragment read
    stage16B(g + 4, l + 4);
  }
  stage_wait();
  __syncthreads();
  const int lane = t & 31, khalf = lane >> 4, mn = lane & 15;
  const int n0 = (t >> 5) * 16;
  v8f acc = {};
  #pragma unroll
  for (int k0 = 0; k0 < F; k0 += 4) {
    int kb = k0 + 2 * khalf;
    v2f a, b;
    a.x = fmaxf(As[mn * 132 + kb], 0.f);
    a.y = fmaxf(As[mn * 132 + kb + 1], 0.f);
    // B[k][n] = w_lin[n][k]  (w_lin is [H,F], we need its transpose)
    b.x = wlin[(size_t)(n0 + mn) * F + kb];
    b.y = wlin[(size_t)(n0 + mn) * F + kb + 1];
    acc = __builtin_amdgcn_wmma_f32_16x16x4_f32(false, a, false, b, (short)0, acc, false, false);
  }
  float bias = blin[n0 + mn];
  #pragma unroll
  for (int r = 0; r < 8; ++r)
    out[(size_t)(m0 + r + 8 * khalf) * F + n0 + mn] = acc[r] + bias;
}

extern "C" void kernel_launch(void* const* d_in, const int* in_sizes, int n_in,
                              void* d_out, int out_size, void* d_ws, size_t ws_size,
                              hipStream_t stream) {
  const float* x     = (const float*)d_in[0];
  const float* ew    = (const float*)d_in[1];
  const float* p     = (const float*)d_in[2];
  const float* w_ih  = (const float*)d_in[3];
  const float* w_hh  = (const float*)d_in[4];
  const float* b_ih  = (const float*)d_in[5];
  const float* b_hh  = (const float*)d_in[6];
  const float* w0    = (const float*)d_in[7];
  const float* w_lin = (const float*)d_in[8];
  const float* b_lin = (const float*)d_in[9];
  const int*   ei    = (const int*)d_in[10];

  const int n = in_sizes[0] / F;   // 100000 (multiple of 16)
  const int E = in_sizes[1];       // 3200000
  float* out = (float*)d_out;

  char* wsb = (char*)d_ws;
  size_t off = 0;
  auto take = [&](size_t bytes) -> void* {
    void* ptr = wsb + off;
    off = (off + bytes + 255) & ~(size_t)255;
    return ptr;
  };
  float* scores = (float*)take((size_t)n * 4);
  float* pn     = (float*)take(4);
  int*   topidx = (int*)take(128 * 4);
  float* topval = (float*)take(128 * 4);
  float* xt     = (float*)take((size_t)F * F * 4);
  float* Wm     = (float*)take((size_t)F * F * 4);
  float* deg    = (float*)take((size_t)n * 4);
  float* dinv   = (float*)take((size_t)n * 4);
  float* xw     = (float*)take((size_t)n * F * 4);
  float* agg    = (float*)take((size_t)n * F * 4);
  (void)ws_size; (void)n_in; (void)out_size;

  pnorm_kernel<<<1, 32, 0, stream>>>(p, pn);
  score_kernel<<<(n + 7) / 8, 256, 0, stream>>>(x, p, pn, scores, n);
  topk_kernel<<<1, 1024, 0, stream>>>(scores, n, topidx, topval);
  gather_kernel<<<F, F, 0, stream>>>(x, topidx, topval, xt);
  gru_kernel<<<F, F, 0, stream>>>(xt, w0, w_ih, w_hh, b_ih, b_hh, Wm);
  zero_kernel<<<(n + 255) / 256, 256, 0, stream>>>(deg, n);
  deg_kernel<<<(int)(((long long)E + n + 255) / 256), 256, 0, stream>>>(ei, ew, deg, E, n);
  dinv_kernel<<<(n + 255) / 256, 256, 0, stream>>>(deg, dinv, n);
  gemm_xw_kernel<<<n / 16, 256, 0, stream>>>(x, Wm, xw);
  agginit_kernel<<<(int)(((size_t)n * F + 255) / 256), 256, 0, stream>>>(xw, dinv, agg, n);
  edge_kernel<<<(E + 7) / 8, 256, 0, stream>>>(ei, ew, dinv, xw, agg, E);
  gemm_out_kernel<<<n / 16, 256, 0, stream>>>(agg, w_lin, b_lin, out);
}